// MessagePassingMultiQuantV2_45492293599395
// MI455X (gfx1250) — compile-verified
//
#include <hip/hip_runtime.h>

typedef __attribute__((ext_vector_type(2))) float v2f;
typedef __attribute__((ext_vector_type(8))) float v8f;
typedef __attribute__((__vector_size__(2 * sizeof(int)))) int v2i_b64;   // async b64 payload type

#define FD 64          // out_features (fixed by reference)
#define CSR_PAD 64     // sentinel pairs appended to CSR stream

#if __has_builtin(__builtin_amdgcn_global_load_async_to_lds_b64) && \
    __has_builtin(__builtin_amdgcn_s_wait_asynccnt)
#define HAVE_ASYNC_LDS 1
#else
#define HAVE_ASYNC_LDS 0
#endif

// ---------- fake quant (LSQ forward): round-half-even matches jnp.round ----------
__device__ __forceinline__ float fq1(float v, float a) {
  float s = v / a;
  s = fminf(fmaxf(s, -8.0f), 7.0f);   // qn=-8, qp=7 (BIT=4)
  return rintf(s) * a;                // RNE
}

__global__ void quant_kernel(const float* __restrict__ x, const float* __restrict__ alpha,
                             float* __restrict__ xq, int total4) {
  int i = blockIdx.x * blockDim.x + threadIdx.x;
  if (i >= total4) return;
  int idx = i * 4;                    // rows of 64 -> feature index = idx & 63, +3 stays in row
  float4 v = *(const float4*)(x + idx);
  int f = idx & (FD - 1);
  float4 r;
  r.x = fq1(v.x, alpha[f + 0]);
  r.y = fq1(v.y, alpha[f + 1]);
  r.z = fq1(v.z, alpha[f + 2]);
  r.w = fq1(v.w, alpha[f + 3]);
  *(float4*)(xq + idx) = r;
}

// ---------- helpers ----------
__global__ void zero_int_kernel(int* __restrict__ p, int n) {
  int i = blockIdx.x * blockDim.x + threadIdx.x;
  if (i < n) p[i] = 0;
}
__global__ void zero_float_kernel(float* __restrict__ p, int n) {
  int i = blockIdx.x * blockDim.x + threadIdx.x;
  if (i < n) p[i] = 0.0f;
}
__global__ void pad_kernel(int2* __restrict__ csr, int E) {
  int i = threadIdx.x;                       // 64 threads
  if (i < CSR_PAD) csr[E + i] = make_int2(0, -1);   // src=0 (safe row), dst=-1 (never matches)
}

// ---------- CSR build: count degrees per destination ----------
__global__ void count_kernel(const int* __restrict__ dst, int* __restrict__ deg, int E) {
  int e = blockIdx.x * blockDim.x + threadIdx.x;
  if (e < E) atomicAdd(&deg[dst[e]], 1);
}

// ---------- single-workgroup exclusive scan: deg(=cursor) -> rowstart, reset cursor ----------
__global__ void scan_kernel(int* __restrict__ deg_cursor, int* __restrict__ rowstart, int n) {
  __shared__ int sm[1024];
  __shared__ int s_carry;
  int tid = threadIdx.x;
  if (tid == 0) s_carry = 0;
  __syncthreads();
  for (int b = 0; b < n; b += 1024) {
    int i = b + tid;
    int v = (i < n) ? deg_cursor[i] : 0;
    sm[tid] = v;
    __syncthreads();
    for (int off = 1; off < 1024; off <<= 1) {     // Hillis-Steele inclusive scan in LDS
      int t = (tid >= off) ? sm[tid - off] : 0;
      __syncthreads();
      sm[tid] += t;
      __syncthreads();
    }
    int carry = s_carry;
    int excl = carry + sm[tid] - v;
    if (i < n) { rowstart[i] = excl; deg_cursor[i] = excl; }  // cursor := rowstart
    __syncthreads();
    if (tid == 1023) s_carry = carry + sm[1023];
    __syncthreads();
  }
  if (tid == 0) rowstart[n] = s_carry;  // == E
}

// ---------- CSR fill (int atomics only), packed (src,dst) pairs ----------
__global__ void fill_kernel(const int* __restrict__ src, const int* __restrict__ dst,
                            int* __restrict__ cursor, int2* __restrict__ csr, int E) {
  int e = blockIdx.x * blockDim.x + threadIdx.x;
  if (e >= E) return;
  int d = dst[e];
  int pos = atomicAdd(&cursor[d], 1);
  csr[pos] = make_int2(src[e], d);
}

// ---------- pull phase: WMMA f32 16x16x4 segmented sum ----------
// One wave owns 16 consecutive destination nodes. For each window of 4 CSR edges:
//   A (16x4) = one-hot membership (dst == base+m), B (4x16) = message features,
//   D(16x16 f32) += A x B  -> exact f32 segment_sum. F=64 -> 4 feature tiles.
// CSR pairs are staged 64-at-a-time into wave-private LDS via async-to-LDS.
// Sentinel padding makes the one-hot compare self-masking: no bounds branches.
__global__ void pull_wmma_kernel(const float* __restrict__ xq, const int* __restrict__ rowstart,
                                 const int2* __restrict__ csr, float* __restrict__ out,
                                 int ngroups, int nnodes) {
#if HAVE_ASYNC_LDS
  __shared__ int2 smem[8 * 64];                    // 64 pairs per wave, 8 waves/block
#endif
  int wave_in_blk = threadIdx.x >> 5;
  int wave = blockIdx.x * 8 + wave_in_blk;
  int lane = threadIdx.x & 31;
  if (wave >= ngroups) return;                     // wave-uniform exit (EXEC all-1 at WMMA)

  int base = wave * 16;
  int top  = base + 16; if (top > nnodes) top = nnodes;
  int e0 = rowstart[base];
  int e1 = rowstart[top];

  bool hi = lane >= 16;
  int  m  = lane & 15;
  int  kA = hi ? 2 : 0;                            // K slot in vector component 0
  int  kB = hi ? 3 : 1;                            // K slot in vector component 1
  int  nodeid = base + m;

#if HAVE_ASYNC_LDS
  int2* buf = smem + wave_in_blk * 64;
#endif

  v8f acc0 = {}; v8f acc1 = {}; v8f acc2 = {}; v8f acc3 = {};

  for (int q = e0; q < e1; q += 64) {
#if HAVE_ASYNC_LDS
    // stage 64 (src,dst) pairs into wave-private LDS; ASYNCcnt orders within this wave
    __builtin_amdgcn_global_load_async_to_lds_b64(
        (__attribute__((address_space(1))) v2i_b64*)(csr + q + lane),
        (__attribute__((address_space(3))) v2i_b64*)(buf + lane), 0, 0);
    __builtin_amdgcn_global_load_async_to_lds_b64(
        (__attribute__((address_space(1))) v2i_b64*)(csr + q + 32 + lane),
        (__attribute__((address_space(3))) v2i_b64*)(buf + 32 + lane), 0, 0);
    __builtin_prefetch(csr + q + 64, 0, 3);        // global_prefetch_b8, near-temporal
    __builtin_amdgcn_s_wait_asynccnt(0);
#else
    __builtin_prefetch(csr + q + 64, 0, 3);
#endif
    int lim = (q + 64 <= e1) ? 64 : (e1 - q);      // windows may straddle e1: sentinel/self-mask
    for (int w = 0; w < lim; w += 4) {
#if HAVE_ASYNC_LDS
      int2 eA = buf[w + kA];                       // ds_load_b64
      int2 eB = buf[w + kB];
#else
      int2 eA = csr[q + w + kA];                   // global_load_b64 (L2-resident)
      int2 eB = csr[q + w + kB];
#endif
      v2f A;                                       // A 16x4: v0={K0|K2}, v1={K1|K3} per lane-half
      A[0] = (eA.y == nodeid) ? 1.0f : 0.0f;
      A[1] = (eB.y == nodeid) ? 1.0f : 0.0f;

      const float* rA = xq + (size_t)eA.x * FD + m;
      const float* rB = xq + (size_t)eB.x * FD + m;
      v2f B0; B0[0] = rA[0];  B0[1] = rB[0];       // B 4x16: N across lanes, K per lane-half
      v2f B1; B1[0] = rA[16]; B1[1] = rB[16];
      v2f B2; B2[0] = rA[32]; B2[1] = rB[32];
      v2f B3; B3[0] = rA[48]; B3[1] = rB[48];

      acc0 = __builtin_amdgcn_wmma_f32_16x16x4_f32(false, A, false, B0, (short)0, acc0, false, false);
      acc1 = __builtin_amdgcn_wmma_f32_16x16x4_f32(false, A, false, B1, (short)0, acc1, false, false);
      acc2 = __builtin_amdgcn_wmma_f32_16x16x4_f32(false, A, false, B2, (short)0, acc2, false, false);
      acc3 = __builtin_amdgcn_wmma_f32_16x16x4_f32(false, A, false, B3, (short)0, acc3, false, false);
    }
  }

  // D 16x16 layout: component r -> M = r (lanes 0-15) / M = 8+r (lanes 16-31), N = lane&15
  #pragma unroll
  for (int r = 0; r < 8; ++r) {
    int row = base + (hi ? 8 + r : r);
    if (row < nnodes) {
      float* orow = out + (size_t)row * FD + m;
      orow[0]  = acc0[r];
      orow[16] = acc1[r];
      orow[32] = acc2[r];
      orow[48] = acc3[r];
    }
  }
}

// ---------- fallback (workspace too small): atomic push with inline quant ----------
__global__ void push_kernel(const float* __restrict__ x, const float* __restrict__ alpha,
                            const int* __restrict__ src, const int* __restrict__ dst,
                            float* __restrict__ out, int E) {
  int gid = blockIdx.x * blockDim.x + threadIdx.x;
  int e = gid >> 5;
  int l = gid & 31;
  if (e >= E) return;
  int s = src[e], d = dst[e];
  int f = l * 2;
  float2 v = *(const float2*)(x + (size_t)s * FD + f);
  atomicAdd(out + (size_t)d * FD + f + 0, fq1(v.x, alpha[f + 0]));
  atomicAdd(out + (size_t)d * FD + f + 1, fq1(v.y, alpha[f + 1]));
}

extern "C" void kernel_launch(void* const* d_in, const int* in_sizes, int n_in,
                              void* d_out, int out_size, void* d_ws, size_t ws_size,
                              hipStream_t stream) {
  const float* x     = (const float*)d_in[0];
  const float* alpha = (const float*)d_in[1];
  const int*   edge  = (const int*)d_in[2];
  int NF = in_sizes[0];
  int nn = NF / FD;                 // nodes
  int E  = in_sizes[2] / 2;         // edges
  const int* src = edge;            // edge_index[0] = sources
  const int* dst = edge + E;        // edge_index[1] = targets
  float* out = (float*)d_out;

  size_t need = (size_t)NF * 4                 // xq
              + (size_t)(E + CSR_PAD) * 8      // packed CSR pairs (+ sentinel pad)
              + (size_t)(nn + 1) * 4           // rowstart
              + (size_t)nn * 4;                // cursor/deg

  if (ws_size >= need) {
    float* xq       = (float*)d_ws;
    int2*  csr      = (int2*)(xq + NF);                  // 8B-aligned: NF*4 divisible by 8
    int*   rowstart = (int*)(csr + (E + CSR_PAD));
    int*   cursor   = rowstart + (nn + 1);

    zero_int_kernel<<<(nn + 255) / 256, 256, 0, stream>>>(cursor, nn);
    pad_kernel<<<1, 64, 0, stream>>>(csr, E);
    quant_kernel<<<((NF / 4) + 255) / 256, 256, 0, stream>>>(x, alpha, xq, NF / 4);
    count_kernel<<<(E + 255) / 256, 256, 0, stream>>>(dst, cursor, E);
    scan_kernel<<<1, 1024, 0, stream>>>(cursor, rowstart, nn);
    fill_kernel<<<(E + 255) / 256, 256, 0, stream>>>(src, dst, cursor, csr, E);

    int ngroups = (nn + 15) / 16;
    int blocks  = (ngroups + 7) / 8;      // 8 waves (of 32) per 256-thread block
    pull_wmma_kernel<<<blocks, 256, 0, stream>>>(xq, rowstart, csr, out, ngroups, nn);
  } else {
    zero_float_kernel<<<(out_size + 255) / 256, 256, 0, stream>>>(out, out_size);
    long long total = (long long)E * 32;
    push_kernel<<<(int)((total + 255) / 256), 256, 0, stream>>>(x, alpha, src, dst, out, E);
  }
}